// GCN_30227979829592
// MI455X (gfx1250) — compile-verified
//
#include <hip/hip_runtime.h>
#include <hip/hip_bf16.h>
#include <math.h>

#define N_NODES   100000
#define N_EDGES   1600000
#define D_IN      128
#define D_HID     64
#define N_CLASSES 40
#define N_GRAPHS  64

typedef __attribute__((ext_vector_type(2))) float v2f;
typedef __attribute__((ext_vector_type(8))) float v8f;

// ---------------- degree / normalization ----------------
__global__ void gcn_deg_init(float* __restrict__ deg) {
  int i = blockIdx.x * blockDim.x + threadIdx.x;
  if (i < N_NODES) deg[i] = 1.0f;               // self-loop contributes 1
}

__global__ void gcn_deg_edges(const int* __restrict__ dst, float* __restrict__ deg) {
  int e = blockIdx.x * blockDim.x + threadIdx.x;
  if (e < N_EDGES) atomicAdd(&deg[dst[e]], 1.0f);
}

__global__ void gcn_deg_rsqrt(float* __restrict__ deg) {
  int i = blockIdx.x * blockDim.x + threadIdx.x;
  if (i < N_NODES) deg[i] = rsqrtf(deg[i]);     // deg_inv_sqrt in place
}

// ---------------- WMMA GEMM: C[M x 64] = act(A[M x K]) @ B[K x 64] ----------------
// 256-thread block = 8 waves; block stages B in LDS (padded stride, bank-clean),
// each wave32 computes a 16x64 strip: one float2 A-load feeds 4 WMMAs.
template <int K, bool RELU>
__global__ __launch_bounds__(256) void gcn_gemm_wmma(const float* __restrict__ A,
                                                     const float* __restrict__ B,
                                                     float* __restrict__ C, int M) {
  constexpr int LDB = 66;                       // pad: rows k, k+1 hit distinct banks
  __shared__ float Bs[K * LDB];
  const int tid = threadIdx.x;
  for (int i = tid; i < K * D_HID; i += 256)
    Bs[(i >> 6) * LDB + (i & 63)] = B[i];
  __syncthreads();

  const int strip = blockIdx.x * 8 + (tid >> 5);   // 16-row output strip per wave
  if (strip >= (M >> 4)) return;
  const int lane = tid & 31, lo = lane & 15, hi = lane >> 4;
  const float* Aptr = A + (size_t)((strip << 4) + lo) * K;  // A: M = lane%16

  v8f acc0 = {}, acc1 = {}, acc2 = {}, acc3 = {};
#pragma unroll
  for (int k = 0; k < K; k += 4) {
    // A 16x4 f32 layout: VGPR0 = K / K+2 per half-wave, VGPR1 = +1 -> contiguous float2
    v2f a = *(const v2f*)(Aptr + k + 2 * hi);
    if (RELU) { a.x = fmaxf(a.x, 0.0f); a.y = fmaxf(a.y, 0.0f); }
    const float* brow = Bs + (k + 2 * hi) * LDB;  // B row k+2*hi; +LDB is row k+2*hi+1
    v2f b;
    b.x = brow[lo];      b.y = brow[LDB + lo];
    acc0 = __builtin_amdgcn_wmma_f32_16x16x4_f32(false, a, false, b, (short)0, acc0, false, false);
    b.x = brow[16 + lo]; b.y = brow[LDB + 16 + lo];
    acc1 = __builtin_amdgcn_wmma_f32_16x16x4_f32(false, a, false, b, (short)0, acc1, false, false);
    b.x = brow[32 + lo]; b.y = brow[LDB + 32 + lo];
    acc2 = __builtin_amdgcn_wmma_f32_16x16x4_f32(false, a, false, b, (short)0, acc2, false, false);
    b.x = brow[48 + lo]; b.y = brow[LDB + 48 + lo];
    acc3 = __builtin_amdgcn_wmma_f32_16x16x4_f32(false, a, false, b, (short)0, acc3, false, false);
  }

#pragma unroll
  for (int r = 0; r < 8; ++r) {                   // D: VGPR r -> rows r / r+8 per half-wave
    float* crow = C + (size_t)((strip << 4) + r + 8 * hi) * D_HID;
    crow[lo]      = acc0[r];
    crow[16 + lo] = acc1[r];
    crow[32 + lo] = acc2[r];
    crow[48 + lo] = acc3[r];
  }
}

// ---------------- self-loop + bias init: out = h * dis^2 + b ----------------
__global__ void gcn_selfbias(const float* __restrict__ h, const float* __restrict__ dis,
                             const float* __restrict__ b, float* __restrict__ out) {
  int gid = blockIdx.x * blockDim.x + threadIdx.x;
  if (gid >= N_NODES * D_HID) return;
  int i = gid >> 6, d = gid & 63;
  float w = dis[i];
  out[gid] = h[gid] * w * w + b[d];
}

// ---------------- edge scatter: out[dst] += h[src] * dis[src]*dis[dst] ----------------
// One thread per (edge, channel); h/out buffers are L2-resident (25.6MB << 192MB L2).
__global__ void gcn_edge(const float* __restrict__ h, const float* __restrict__ dis,
                         const int* __restrict__ src, const int* __restrict__ dst,
                         float* __restrict__ out) {
  int gid = blockIdx.x * blockDim.x + threadIdx.x;
  if (gid >= N_EDGES * D_HID) return;
  int e = gid >> 6, d = gid & 63;
  int s = src[e], t = dst[e];
  float w = dis[s] * dis[t];
  atomicAdd(&out[(size_t)t * D_HID + d], h[(size_t)s * D_HID + d] * w);
}

// ---------------- pooling ----------------
__global__ void gcn_zero(float* __restrict__ p, int n) {
  int i = blockIdx.x * blockDim.x + threadIdx.x;
  if (i < n) p[i] = 0.0f;
}

__global__ void gcn_pool(const float* __restrict__ h, const int* __restrict__ batch,
                         float* __restrict__ sums, float* __restrict__ counts) {
  int gid = blockIdx.x * blockDim.x + threadIdx.x;
  if (gid >= N_NODES * D_HID) return;
  int i = gid >> 6, d = gid & 63;
  int g = batch[i];
  atomicAdd(&sums[g * D_HID + d], h[gid]);
  if (d == 0) atomicAdd(&counts[g], 1.0f);
}

// ---------------- MLP head + log_softmax (one block per graph) ----------------
__global__ void gcn_mlp(const float* __restrict__ sums, const float* __restrict__ counts,
                        const float* __restrict__ Wm1, const float* __restrict__ bm1,
                        const float* __restrict__ Wm2, const float* __restrict__ bm2,
                        float* __restrict__ out) {
  __shared__ float row[D_HID], tbuf[D_HID], zred[N_CLASSES], stat[2];
  int g = blockIdx.x, d = threadIdx.x;
  float cnt = counts[g];
  cnt = cnt > 1.0f ? cnt : 1.0f;
  row[d] = sums[g * D_HID + d] / cnt;
  __syncthreads();
  float t = bm1[d];
  for (int k = 0; k < D_HID; ++k) t += row[k] * Wm1[k * D_HID + d];
  tbuf[d] = t > 0.0f ? t : 0.0f;
  __syncthreads();
  if (d < N_CLASSES) {
    float z = bm2[d];
    for (int k = 0; k < D_HID; ++k) z += tbuf[k] * Wm2[k * N_CLASSES + d];
    zred[d] = z;
  }
  __syncthreads();
  if (d == 0) {
    float m = zred[0];
    for (int c = 1; c < N_CLASSES; ++c) m = fmaxf(m, zred[c]);
    float s = 0.0f;
    for (int c = 0; c < N_CLASSES; ++c) s += expf(zred[c] - m);
    stat[0] = m;
    stat[1] = logf(s);
  }
  __syncthreads();
  if (d < N_CLASSES) out[g * N_CLASSES + d] = zred[d] - stat[0] - stat[1];
}

// ---------------- driver ----------------
extern "C" void kernel_launch(void* const* d_in, const int* in_sizes, int n_in,
                              void* d_out, int out_size, void* d_ws, size_t ws_size,
                              hipStream_t stream) {
  (void)in_sizes; (void)n_in; (void)out_size; (void)ws_size;

  const float* x    = (const float*)d_in[0];
  const int*   ei   = (const int*)  d_in[1];
  const int*   bat  = (const int*)  d_in[2];
  const float* W1   = (const float*)d_in[3];
  const float* b1   = (const float*)d_in[4];
  const float* W2   = (const float*)d_in[5];
  const float* b2   = (const float*)d_in[6];
  const float* W3   = (const float*)d_in[7];
  const float* b3   = (const float*)d_in[8];
  const float* Wm1  = (const float*)d_in[9];
  const float* bm1  = (const float*)d_in[10];
  const float* Wm2  = (const float*)d_in[11];
  const float* bm2  = (const float*)d_in[12];
  const int* src = ei;
  const int* dst = ei + N_EDGES;

  float* ws     = (float*)d_ws;
  float* dis    = ws;                                      // 100000 (padded to 100096)
  float* H      = ws + 100096;                             // 6.4M : X@W product
  float* AG     = H  + (size_t)N_NODES * D_HID;            // 6.4M : aggregated layer output
  float* pooled = AG + (size_t)N_NODES * D_HID;            // 64*64
  float* counts = pooled + N_GRAPHS * D_HID;               // 64

  const int TPB    = 256;
  const int nNV    = (N_NODES * D_HID + TPB - 1) / TPB;    // node*channel grid
  const int nEV    = (N_EDGES * D_HID + TPB - 1) / TPB;    // edge*channel grid
  const int strips = N_NODES / 16;                          // 6250 wave-strips
  const int gemmB  = (strips + 7) / 8;                      // 8 waves per block

  // degree -> deg_inv_sqrt
  gcn_deg_init <<<(N_NODES + TPB - 1) / TPB, TPB, 0, stream>>>(dis);
  gcn_deg_edges<<<(N_EDGES + TPB - 1) / TPB, TPB, 0, stream>>>(dst, dis);
  gcn_deg_rsqrt<<<(N_NODES + TPB - 1) / TPB, TPB, 0, stream>>>(dis);

  // layer 1: h = x@W1 ; agg = scatter + self + b1   (relu fused into next GEMM)
  gcn_gemm_wmma<D_IN, false><<<gemmB, TPB, 0, stream>>>(x, W1, H, N_NODES);
  gcn_selfbias<<<nNV, TPB, 0, stream>>>(H, dis, b1, AG);
  gcn_edge    <<<nEV, TPB, 0, stream>>>(H, dis, src, dst, AG);

  // layer 2: h = relu(AG)@W2
  gcn_gemm_wmma<D_HID, true><<<gemmB, TPB, 0, stream>>>(AG, W2, H, N_NODES);
  gcn_selfbias<<<nNV, TPB, 0, stream>>>(H, dis, b2, AG);
  gcn_edge    <<<nEV, TPB, 0, stream>>>(H, dis, src, dst, AG);

  // layer 3: h = relu(AG)@W3 (no relu on layer output)
  gcn_gemm_wmma<D_HID, true><<<gemmB, TPB, 0, stream>>>(AG, W3, H, N_NODES);
  gcn_selfbias<<<nNV, TPB, 0, stream>>>(H, dis, b3, AG);
  gcn_edge    <<<nEV, TPB, 0, stream>>>(H, dis, src, dst, AG);

  // pool + head
  gcn_zero<<<(N_GRAPHS * D_HID + N_GRAPHS + TPB - 1) / TPB, TPB, 0, stream>>>(
      pooled, N_GRAPHS * D_HID + N_GRAPHS);
  gcn_pool<<<nNV, TPB, 0, stream>>>(AG, bat, pooled, counts);
  gcn_mlp <<<N_GRAPHS, D_HID, 0, stream>>>(pooled, counts, Wm1, bm1, Wm2, bm2,
                                           (float*)d_out);
}